// DetContrastiveLoss_72636486910298
// MI455X (gfx1250) — compile-verified
//
#include <hip/hip_runtime.h>
#include <math.h>

typedef __attribute__((ext_vector_type(2))) float v2f;
typedef __attribute__((ext_vector_type(8))) float v8f;

#define CCH 128          // channels (K)
#define NPAD 512         // padded N for 16x16 tiling
#define TEMP_INV 10.0f   // 1 / TEMPERATURE

// ---------------- Kernel A: gather features at box cells + double-normalize rows
__global__ void gather_norm_kernel(const float* __restrict__ sf, const float* __restrict__ gt,
                                   float* __restrict__ f, int B, int N, int H, int W) {
    int n = blockIdx.x;        // 0..NPAD-1 (uniform per block)
    int c = threadIdx.x;       // 0..127
    __shared__ float red[CCH];
    __shared__ int s_idx;
    float v = 0.0f;
    if (n < N) {
        if (c == 0) {
            float x = gt[((size_t)(B - 1) * N + n) * 8 + 0];
            float y = gt[((size_t)(B - 1) * N + n) * 8 + 1];
            int cx = (int)((x + 59.9f) / 119.8f * (float)W);
            int cy = (int)((y + 59.9f) / 119.8f * (float)H);
            cx = min(max(cx, 0), W - 1);
            cy = min(max(cy, 0), H - 1);
            s_idx = cy * W + cx;
        }
        __syncthreads();
        size_t base = (size_t)(B - 1) * CCH * (size_t)H * W;
        v = sf[base + (size_t)c * H * W + s_idx];
    } else {
        __syncthreads();  // keep barrier count uniform (n is block-uniform)
    }
    red[c] = v * v;
    __syncthreads();
    for (int s = CCH / 2; s > 0; s >>= 1) {
        if (c < s) red[c] += red[c + s];
        __syncthreads();
    }
    float s1 = sqrtf(red[0]);
    float n1 = fmaxf(s1, 1e-12f);
    float s2 = s1 / n1;                 // norm of the once-normalized vector
    float n2 = fmaxf(s2, 1e-8f);
    float scale = 1.0f / (n1 * n2);
    f[(size_t)n * CCH + c] = (n < N) ? v * scale : 0.0f;   // zero pad rows
}

// ---------------- Kernel B: sim = (f @ f^T) * (1/T) via V_WMMA_F32_16X16X4_F32
// One wave computes one 16x16 output tile; K=128 -> 32 chained WMMAs.
__global__ void sim_wmma_kernel(const float* __restrict__ f, float* __restrict__ sim) {
    int wid  = (int)((blockIdx.x * blockDim.x + threadIdx.x) >> 5);
    int lane = (int)(threadIdx.x & 31);
    int tr = wid >> 5;          // tile row 0..31
    int tc = wid & 31;          // tile col 0..31
    int half = lane >> 4;       // 0: K pair {0,1}, 1: K pair {2,3}
    int m = lane & 15;

    const float* arow = f + (size_t)(tr * 16 + m) * CCH;   // A[M=m][K...]
    const float* brow = f + (size_t)(tc * 16 + m) * CCH;   // B[K][N=m] = f[col][K]

    v8f acc = {};
    for (int k = 0; k < CCH; k += 4) {
        int kk = k + 2 * half;
        v2f a = *(const v2f*)(arow + kk);
        v2f b = *(const v2f*)(brow + kk);
        acc = __builtin_amdgcn_wmma_f32_16x16x4_f32(
            /*neg_a=*/false, a, /*neg_b=*/false, b,
            /*c_mod=*/(short)0, acc, /*reuse_a=*/false, /*reuse_b=*/false);
    }

    int col = tc * 16 + m;
    int rowBase = tr * 16 + 8 * half;
#pragma unroll
    for (int r = 0; r < 8; ++r) {
        sim[(size_t)(rowBase + r) * NPAD + col] = acc[r] * TEMP_INV;
    }
}

// ---------------- Kernel C: per-row log-softmax over valid columns, pick label
__global__ void rowloss_kernel(const float* __restrict__ sim, const int* __restrict__ slab,
                               const int* __restrict__ dlab, const int* __restrict__ nstat,
                               float* __restrict__ rowloss, int N) {
    int i = blockIdx.x;
    int tid = threadIdx.x;
    __shared__ float red[256];
    const float* row = sim + (size_t)i * NPAD;

    float lmax = -INFINITY;
    for (int j = tid; j < N; j += 256) lmax = fmaxf(lmax, row[j]);
    red[tid] = lmax; __syncthreads();
    for (int s = 128; s > 0; s >>= 1) {
        if (tid < s) red[tid] = fmaxf(red[tid], red[tid + s]);
        __syncthreads();
    }
    float rmax = red[0];
    __syncthreads();

    float lsum = 0.0f;
    for (int j = tid; j < N; j += 256) lsum += expf(row[j] - rmax);
    red[tid] = lsum; __syncthreads();
    for (int s = 128; s > 0; s >>= 1) {
        if (tid < s) red[tid] += red[tid + s];
        __syncthreads();
    }
    if (tid == 0) {
        int Ns = nstat[0];
        int lab = (i < Ns) ? slab[i] : dlab[i - Ns];
        rowloss[i] = -(row[lab] - rmax - logf(red[0]));
    }
}

// ---------------- Kernel D: mean over rows, scale, write scalar
__global__ void final_reduce_kernel(const float* __restrict__ rowloss, float* __restrict__ out, int N) {
    __shared__ float red[256];
    int tid = threadIdx.x;
    float s = 0.0f;
    for (int j = tid; j < N; j += 256) s += rowloss[j];
    red[tid] = s; __syncthreads();
    for (int t = 128; t > 0; t >>= 1) {
        if (tid < t) red[tid] += red[tid + t];
        __syncthreads();
    }
    if (tid == 0) out[0] = red[0] / (float)N * 0.01f;
}

extern "C" void kernel_launch(void* const* d_in, const int* in_sizes, int n_in,
                              void* d_out, int out_size, void* d_ws, size_t ws_size,
                              hipStream_t stream) {
    const float* sf   = (const float*)d_in[0];   // (B,128,512,512) f32
    const float* gt   = (const float*)d_in[1];   // (B,N,8) f32
    const int*  slab  = (const int*)d_in[2];     // (Ns,) i32
    const int*  dlab  = (const int*)d_in[3];     // (Nd,) i32
    const int*  nstat = (const int*)d_in[4];     // scalar i32

    const int C = CCH, H = 512, W = 512;
    int N = in_sizes[2] + in_sizes[3];           // 500
    int B = in_sizes[0] / (C * H * W);           // 4

    float* f       = (float*)d_ws;               // NPAD * C
    float* sim     = f + (size_t)NPAD * C;       // NPAD * NPAD
    float* rowloss = sim + (size_t)NPAD * NPAD;  // NPAD

    gather_norm_kernel<<<NPAD, CCH, 0, stream>>>(sf, gt, f, B, N, H, W);

    // (512/16)^2 = 1024 tiles, 1 wave (32 threads) per tile, 8 waves per 256-thread block
    int tiles = (NPAD / 16) * (NPAD / 16);
    sim_wmma_kernel<<<tiles / 8, 256, 0, stream>>>(f, sim);

    rowloss_kernel<<<N, 256, 0, stream>>>(sim, slab, dlab, nstat, rowloss, N);
    final_reduce_kernel<<<1, 256, 0, stream>>>(rowloss, (float*)d_out, N);
}